// HT2SPHERE_77163382440038
// MI455X (gfx1250) — compile-verified
//
#include <hip/hip_runtime.h>

typedef __attribute__((ext_vector_type(2))) float v2f;
typedef __attribute__((ext_vector_type(8))) float v8f;

#define BC 256       // B*C rows of flat
#define HW 16384     // H*W
#define SS 16384     // sphere size

// ---------------------------------------------------------------------------
// Generic fp32 16x16-tile transpose via V_WMMA_F32_16X16X4_F32.
//   src: (SR x SC) row-major,  dst: (SC x SR) row-major,  dst[q,p] = src[p,q]
// Trick: D = A x I. A-operand 16x4 f32 layout (ISA 7.12.2): lane = m + 16*h,
// vgpr j holds K = j + 2*h. Feeding A(m,kk) = src[a0+4k+kk][b0+m] and
// B = rows 4k..4k+3 of identity gives D(m,n) = src[a0+n][b0+m], i.e. the
// transposed tile, already in the D layout (M across VGPRs, N across lanes)
// so the store to dst is coalesced per half-wave. Exact fp32, 4 WMMAs/tile.
// Requires SR, SC multiples of 16 and EXEC all-ones (grid sized exactly).
// ---------------------------------------------------------------------------
__global__ void wmma_transpose_f32(const float* __restrict__ src,
                                   float* __restrict__ dst,
                                   int SR, int SC) {
    const int lane  = threadIdx.x & 31;
    const int wave  = threadIdx.x >> 5;
    const int tile  = blockIdx.x * (blockDim.x >> 5) + wave;
    const int tilesA = SR >> 4;              // tiles along src rows
    const int a0 = (tile % tilesA) << 4;     // src row origin
    const int b0 = (tile / tilesA) << 4;     // src col origin
    const int m  = lane & 15;
    const int h  = lane >> 4;                // half-wave select

    v8f c = {};
#pragma unroll
    for (int k = 0; k < 4; ++k) {
        const int ka = 4 * k + 2 * h;        // absolute K index for vgpr 0
        v2f a, b;
        a.x = src[(size_t)(a0 + ka + 0) * SC + b0 + m];
        a.y = src[(size_t)(a0 + ka + 1) * SC + b0 + m];
        b.x = (m == ka + 0) ? 1.0f : 0.0f;   // identity rows 4k..4k+3
        b.y = (m == ka + 1) ? 1.0f : 0.0f;
        c = __builtin_amdgcn_wmma_f32_16x16x4_f32(false, a, false, b,
                                                  (short)0, c, false, false);
    }
#pragma unroll
    for (int j = 0; j < 8; ++j) {
        // c[j]: M = j + 8*h, N = m  ->  dst[b0 + M][a0 + N]
        dst[(size_t)(b0 + j + 8 * h) * SR + a0 + m] = c[j];
    }
}

// ---------------------------------------------------------------------------
// Zero the transposed accumulator (ws is poisoned by the harness).
// ---------------------------------------------------------------------------
__global__ void zero_f32(float* __restrict__ p, size_t n) {
    size_t i      = (size_t)blockIdx.x * blockDim.x + threadIdx.x;
    size_t stride = (size_t)gridDim.x * blockDim.x;
    for (; i < n; i += stride) p[i] = 0.0f;
}

// ---------------------------------------------------------------------------
// Vote scatter: each block-iteration handles 4 consecutive votes; the 256
// threads cover the 256 (b*c) rows. Per vote: one fully-coalesced 1 KB
// flatT read and one fully-coalesced 1 KB global_atomic_add_f32 burst into
// outT. Processing 4 votes per iteration gives 4 independent load->atomic
// chains per wave (MLP), instead of one serial chain. vm addresses are
// wave-uniform -> scalar loads. Both arrays are L2-resident (16 MiB each,
// 192 MB L2), so this runs at L2 atomic throughput.
// ---------------------------------------------------------------------------
__global__ void vote_scatter(const float* __restrict__ vm,     // (V,3)
                             const float* __restrict__ flatT,  // (HW, BC)
                             float* __restrict__ outT,         // (SS, BC)
                             int nv) {
    const int    r      = threadIdx.x;            // row 0..255
    const size_t stride = (size_t)gridDim.x * 4;

    for (size_t base = (size_t)blockIdx.x * 4; base < (size_t)nv; base += stride) {
        if (base + 4 <= (size_t)nv) {
            // Fast path: 4 independent votes, no divergence.
            float fh[4], w[4], fs[4];
#pragma unroll
            for (int i = 0; i < 4; ++i) {
                fh[i] = vm[3 * (base + i) + 0];
                w[i]  = vm[3 * (base + i) + 1];
                fs[i] = vm[3 * (base + i) + 2];
            }
            float val[4];
#pragma unroll
            for (int i = 0; i < 4; ++i) {
                val[i] = flatT[(size_t)(int)fh[i] * BC + r] * w[i];
            }
#pragma unroll
            for (int i = 0; i < 4; ++i) {
                atomicAdd(&outT[(size_t)(int)fs[i] * BC + r], val[i]);
            }
        } else {
            for (size_t v = base; v < (size_t)nv; ++v) {
                const float fh = vm[3 * v + 0];
                const float w  = vm[3 * v + 1];
                const float fs = vm[3 * v + 2];
                const float a  = flatT[(size_t)(int)fh * BC + r] * w;
                atomicAdd(&outT[(size_t)(int)fs * BC + r], a);
            }
        }
    }
}

// ---------------------------------------------------------------------------
extern "C" void kernel_launch(void* const* d_in, const int* in_sizes, int n_in,
                              void* d_out, int out_size, void* d_ws, size_t ws_size,
                              hipStream_t stream) {
    const float* x  = (const float*)d_in[0];  // (4,64,128,128) == (256, 16384)
    const float* vm = (const float*)d_in[1];  // (V, 3) float32
    // d_in[2] = sphere_size scalar; fixed at 16384 here.
    float* out   = (float*)d_out;             // (256, 16384)
    float* flatT = (float*)d_ws;              // (16384, 256)  16 MiB
    float* outT  = flatT + (size_t)HW * BC;   // (16384, 256)  16 MiB
    const int nv = in_sizes[1] / 3;

    // 1) flat (256,16384) -> flatT (16384,256): 16384 tiles, 8 waves/block.
    wmma_transpose_f32<<<(BC / 16) * (HW / 16) / 8, 256, 0, stream>>>(
        x, flatT, BC, HW);

    // 2) zero the accumulator.
    zero_f32<<<2048, 256, 0, stream>>>(outT, (size_t)SS * BC);

    // 3) scatter-add all votes, 4 per block-iteration.
    vote_scatter<<<8192, 256, 0, stream>>>(vm, flatT, outT, nv);

    // 4) outT (16384,256) -> out (256,16384).
    wmma_transpose_f32<<<(SS / 16) * (BC / 16) / 8, 256, 0, stream>>>(
        outT, out, SS, BC);
}